// QuantumIQPImageEmbedder_75617194213887
// MI455X (gfx1250) — compile-verified
//
#include <hip/hip_runtime.h>
#include <math.h>

#define NB 4
#define QPB 20
#define NN (1 << 20)
#define INV_SQRT2f 0.70710678118654752f

typedef __attribute__((ext_vector_type(2))) float v2f;
typedef __attribute__((ext_vector_type(8))) float v8f;

__device__ __forceinline__ float2 cmul(float2 a, float2 b) {
  return make_float2(a.x * b.x - a.y * b.y, a.x * b.y + a.y * b.x);
}

// Async global->LDS copy of 16 bytes (ASYNCcnt-tracked, CDNA5).
// ldsByteAddr: LDS byte offset (low 32 bits of generic pointer);
// gaddr: 64-bit global byte address.
__device__ __forceinline__ void async_g2l_b128(unsigned int ldsByteAddr, unsigned long long gaddr) {
  asm volatile("global_load_async_to_lds_b128 %0, %1, off"
               :: "v"(ldsByteAddr), "v"(gaddr) : "memory");
}
__device__ __forceinline__ void async_l2g_b128(unsigned long long gaddr, unsigned int ldsByteAddr) {
  asm volatile("global_store_async_from_lds_b128 %0, %1, off"
               :: "v"(gaddr), "v"(ldsByteAddr) : "memory");
}
__device__ __forceinline__ void wait_asynccnt0() {
  asm volatile("s_wait_asynccnt 0x0" ::: "memory");
}

// ---------------------------------------------------------------------------
// Kernel A: single-qubit states (B,4,20,2) c64 and the four 8x8 Heisenberg ops
// ---------------------------------------------------------------------------
__global__ void k_setup(const float* __restrict__ x, const float* __restrict__ posw,
                        const float* __restrict__ l2w,
                        float2* __restrict__ states, float2* __restrict__ heis) {
  int t = threadIdx.x;
  if (t < 640) {
    int b = t / 80, rem = t % 80, patch = rem / 5, w = rem % 5;
    float ax, ay, az;
    if (w < 4) {
      const float* xp = x + ((b * 16 + patch) * 4 + w) * 3;
      ax = xp[0]; ay = xp[1]; az = xp[2];
    } else {
      ax = posw[patch * 2 + 0]; ay = posw[patch * 2 + 1]; az = 0.f;
    }
    float hx = 0.5f * ax, hy = 0.5f * ay, hz = 0.5f * az;
    float cx = cosf(hx), sx = sinf(hx), cy = cosf(hy), sy = sinf(hy);
    float cz = cosf(hz), sz = sinf(hz);
    // w0 = (cy*cx, sy*sx), w1 = (sy*cx, -cy*sx); then z-phase
    float w0r = cy * cx, w0i = sy * sx, w1r = sy * cx, w1i = -cy * sx;
    float u0r = w0r * cz + w0i * sz, u0i = w0i * cz - w0r * sz;
    float u1r = w1r * cz - w1i * sz, u1i = w1i * cz + w1r * sz;
    int k = patch >> 2, q = (patch & 3) * 5 + w;
    int base = ((b * NB + k) * QPB + q) * 2;
    states[base + 0] = make_float2((u0r + u1r) * INV_SQRT2f, (u0i + u1i) * INV_SQRT2f);
    states[base + 1] = make_float2((u0r - u1r) * INV_SQRT2f, (u0i - u1i) * INV_SQRT2f);
  }
  // Heisenberg ops: H_k = IZI * Rot(a2k,IXI) * Rot(b2[k-1],XXI) * Rot(b2k,IXX)
  __shared__ float2 M1s[4 * 64];
  __shared__ float2 Ps[4 * 64];
  int k = t >> 6, idx = t & 63, i = idx >> 3, j = idx & 7;
  if (t < 256) {
    float b2k = l2w[k * 2 + 1];
    float cb = cosf(b2k), sb = sinf(b2k);
    float2 m = make_float2(0.f, 0.f);
    if (i == j) m.x = cb;               // Rot(b2k, IXX): flip bits {1,0} -> i^3
    if (j == (i ^ 3)) m.y = -sb;
    M1s[k * 64 + idx] = m;
  }
  __syncthreads();
  if (t < 256) {
    float b2p = l2w[((k + 15) & 15) * 2 + 1];
    float cb = cosf(b2p), sb = sinf(b2p);
    float2 m  = M1s[k * 64 + i * 8 + j];
    float2 m2 = M1s[k * 64 + (i ^ 6) * 8 + j];  // XXI: flip bits {2,1}
    Ps[k * 64 + idx] = make_float2(cb * m.x + sb * m2.y, cb * m.y - sb * m2.x);
  }
  __syncthreads();
  if (t < 256) {
    float a2k = l2w[k * 2 + 0];
    float ca = cosf(a2k), sa = sinf(a2k);
    float2 p  = Ps[k * 64 + i * 8 + j];
    float2 p2 = Ps[k * 64 + (i ^ 2) * 8 + j];   // IXI: flip bit 1
    float2 h = make_float2(ca * p.x + sa * p2.y, ca * p.y - sa * p2.x);
    float sgn = (i & 2) ? -1.f : 1.f;           // IZI diagonal
    heis[k * 64 + idx] = make_float2(sgn * h.x, sgn * h.y);
  }
}

__global__ void k_zero(float* __restrict__ p, int n) {
  int i = blockIdx.x * blockDim.x + threadIdx.x;
  if (i < n) p[i] = 0.f;
}

// ---------------------------------------------------------------------------
// Kernel B: psi = (hi-table x lo-table x cross-phase), then 10-stage lo-WHT in
// LDS; one workgroup handles 4 hi values of one slice.
// ---------------------------------------------------------------------------
__global__ void k_psi_lo(const float2* __restrict__ states, const float* __restrict__ l1w,
                         float* __restrict__ psiRe, float* __restrict__ psiIm,
                         int sliceBase) {
  __shared__ float2 S[40];
  __shared__ float L1[40];
  __shared__ float2 Tlo[1024];
  __shared__ float2 buf[1024];
  int t = threadIdx.x;
  int slice = sliceBase + blockIdx.y;
  size_t plane = (size_t)blockIdx.y * NN;
  if (t < 40) { S[t] = states[slice * 40 + t]; L1[t] = l1w[t]; }
  __syncthreads();
  // lo table: qubits 10..19 live in bits 9..0 (qubit q <-> bit 19-q)
  for (int r = 0; r < 4; r++) {
    int l = t + 256 * r;
    float2 prod = make_float2(1.f, 0.f);
    for (int p = 0; p < 10; p++) {
      int q = 19 - p;
      prod = cmul(prod, S[q * 2 + ((l >> p) & 1)]);
    }
    float th = 0.f, zprev = 0.f;
    for (int q = 10; q < 20; q++) {
      float zq = 1.f - 2.f * (float)((l >> (19 - q)) & 1);
      th += L1[q * 2] * zq;
      if (q > 10) th += L1[(q - 1) * 2 + 1] * zprev * zq;
      zprev = zq;
    }
    float ph = 0.5f * th;
    Tlo[l] = cmul(prod, make_float2(cosf(ph), -sinf(ph)));
  }
  __syncthreads();
  for (int hh = 0; hh < 4; hh++) {
    int h = blockIdx.x * 4 + hh;
    float2 prod = make_float2(1.f, 0.f);
    for (int jb = 0; jb < 10; jb++) {
      int q = 9 - jb;
      prod = cmul(prod, S[q * 2 + ((h >> jb) & 1)]);
    }
    float th = 0.f, zprev = 0.f;
    for (int q = 0; q < 10; q++) {
      float zq = 1.f - 2.f * (float)((h >> (9 - q)) & 1);
      th += L1[q * 2] * zq;
      if (q > 0) th += L1[(q - 1) * 2 + 1] * zprev * zq;
      zprev = zq;
    }
    float ph = 0.5f * th;
    float2 Thi = cmul(prod, make_float2(cosf(ph), -sinf(ph)));
    // cross couplings b1[9]*z9*z10 and b1[19]*z19*z0 (bits 10,9,0,19)
    float z9 = 1.f - 2.f * (float)(h & 1);
    float z0 = 1.f - 2.f * (float)((h >> 9) & 1);
    float2 cross[4];
    for (int cc = 0; cc < 4; cc++) {
      float z10 = 1.f - 2.f * (float)(cc >> 1);
      float z19 = 1.f - 2.f * (float)(cc & 1);
      float phc = 0.5f * (L1[9 * 2 + 1] * z9 * z10 + L1[19 * 2 + 1] * z19 * z0);
      cross[cc] = make_float2(cosf(phc), -sinf(phc));
    }
    for (int r = 0; r < 4; r++) {
      int l = t + 256 * r;
      int cc = (((l >> 9) & 1) << 1) | (l & 1);
      buf[l] = cmul(cmul(Tlo[l], Thi), cross[cc]);
    }
    __syncthreads();
    // 10-stage unnormalized WHT over 1024 lo entries
    for (int s = 0; s < 10; s++) {
      for (int r = 0; r < 2; r++) {
        int pid = t + 256 * r;
        int low = pid & ((1 << s) - 1);
        int high = pid >> s;
        int i0 = (high << (s + 1)) | low;
        int i1 = i0 | (1 << s);
        float2 a = buf[i0], bb = buf[i1];
        buf[i0] = make_float2(a.x + bb.x, a.y + bb.y);
        buf[i1] = make_float2(a.x - bb.x, a.y - bb.y);
      }
      __syncthreads();
    }
    for (int r = 0; r < 4; r++) {
      int l = t + 256 * r;
      psiRe[plane + (size_t)h * 1024 + l] = buf[l].x;
      psiIm[plane + (size_t)h * 1024 + l] = buf[l].y;
    }
    __syncthreads();
  }
}

// ---------------------------------------------------------------------------
// Kernel C: hi-axis WHT (stride 1024), 8 lo-columns per workgroup, 64KB LDS.
// Tile staged with GLOBAL_LOAD_ASYNC_TO_LDS_B128 and drained with
// GLOBAL_STORE_ASYNC_FROM_LDS_B128 (ASYNCcnt). Applies the 2^-10 WHT scale.
// ---------------------------------------------------------------------------
__global__ void k_wht_hi(float* __restrict__ psiRe, float* __restrict__ psiIm) {
  __shared__ __align__(16) float Mre[8192];
  __shared__ __align__(16) float Mim[8192];
  int t = threadIdx.x;
  size_t plane = (size_t)blockIdx.y * NN;
  int lo0 = blockIdx.x * 8;
  // Async stage-in: 2048 x 16B chunks per plane. flat = hi*2 + half.
  for (int it = 0; it < 8; it++) {
    int flat = it * 256 + t;
    int half = flat & 1, hi = flat >> 1;
    int e = hi * 8 + half * 4;                     // 4 floats per chunk
    size_t g = plane + (size_t)hi * 1024 + lo0 + half * 4;
    async_g2l_b128((unsigned int)(uintptr_t)&Mre[e],
                   (unsigned long long)(uintptr_t)(psiRe + g));
    async_g2l_b128((unsigned int)(uintptr_t)&Mim[e],
                   (unsigned long long)(uintptr_t)(psiIm + g));
  }
  wait_asynccnt0();
  __syncthreads();
  for (int s = 0; s < 10; s++) {
    for (int u = 0; u < 16; u++) {
      int pc = u * 256 + t;
      int c = pc & 7, pid = pc >> 3;
      int low = pid & ((1 << s) - 1), high = pid >> s;
      int i0 = ((high << (s + 1)) | low) * 8 + c;
      int i1 = i0 + (8 << s);
      float ar = Mre[i0], ai = Mim[i0];
      float br = Mre[i1], bi = Mim[i1];
      Mre[i0] = ar + br; Mim[i0] = ai + bi;
      Mre[i1] = ar - br; Mim[i1] = ai - bi;
    }
    __syncthreads();
  }
  // scale by (1/sqrt2)^20 = 2^-10, then async stage-out
  const float sc = 1.f / 1024.f;
  for (int it = 0; it < 32; it++) {
    int f = it * 256 + t;
    Mre[f] *= sc;
    Mim[f] *= sc;
  }
  __syncthreads();
  for (int it = 0; it < 8; it++) {
    int flat = it * 256 + t;
    int half = flat & 1, hi = flat >> 1;
    int e = hi * 8 + half * 4;
    size_t g = plane + (size_t)hi * 1024 + lo0 + half * 4;
    async_l2g_b128((unsigned long long)(uintptr_t)(psiRe + g),
                   (unsigned int)(uintptr_t)&Mre[e]);
    async_l2g_b128((unsigned long long)(uintptr_t)(psiIm + g),
                   (unsigned int)(uintptr_t)&Mim[e]);
  }
  wait_asynccnt0();
}

// ---------------------------------------------------------------------------
// Kernel D: rho4 = A * A^H per slice via V_WMMA_F32_16X16X4_F32.
// A = psi reshaped (16, 65536); four real Gram GEMMs. A/B register images are
// identical for this symmetric product under the CDNA5 f32 16x4/4x16 layouts:
// lane L<16 holds cols m0..m0+1 of row L, lane 16+L holds cols m0+2..m0+3.
// Manually unrolled x4: 8 b64 loads in flight per trip, then 16 WMMAs.
// ---------------------------------------------------------------------------
__global__ void k_rho(const float* __restrict__ psiRe, const float* __restrict__ psiIm,
                      float* __restrict__ rhoRe, float* __restrict__ rhoIm,
                      int sliceBase) {
  int t = threadIdx.x;
  int lane = t & 31, wave = t >> 5;
  int slice = sliceBase + blockIdx.y;
  size_t plane = (size_t)blockIdx.y * NN;
  int row = lane & 15;
  int cOff = (lane >> 4) * 2;
  int waveG = blockIdx.x * 8 + wave;   // 128 waves/slice, 512 cols each
  int mStart = waveG * 512;
  const float* re = psiRe + plane + (size_t)row * 65536 + mStart + cOff;
  const float* im = psiIm + plane + (size_t)row * 65536 + mStart + cOff;
  v8f accRR = {}, accII = {}, accIR = {}, accRI = {};
  for (int it = 0; it < 32; ++it) {
    int o = it * 16;
    // issue all 8 loads for a 16-column strip first (8 loads in flight)
    v2f ar0 = *(const v2f*)(re + o +  0);
    v2f ai0 = *(const v2f*)(im + o +  0);
    v2f ar1 = *(const v2f*)(re + o +  4);
    v2f ai1 = *(const v2f*)(im + o +  4);
    v2f ar2 = *(const v2f*)(re + o +  8);
    v2f ai2 = *(const v2f*)(im + o +  8);
    v2f ar3 = *(const v2f*)(re + o + 12);
    v2f ai3 = *(const v2f*)(im + o + 12);
    accRR = __builtin_amdgcn_wmma_f32_16x16x4_f32(false, ar0, false, ar0, (short)0, accRR, false, false);
    accII = __builtin_amdgcn_wmma_f32_16x16x4_f32(false, ai0, false, ai0, (short)0, accII, false, false);
    accIR = __builtin_amdgcn_wmma_f32_16x16x4_f32(false, ai0, false, ar0, (short)0, accIR, false, false);
    accRI = __builtin_amdgcn_wmma_f32_16x16x4_f32(false, ar0, false, ai0, (short)0, accRI, false, false);
    accRR = __builtin_amdgcn_wmma_f32_16x16x4_f32(false, ar1, false, ar1, (short)0, accRR, false, false);
    accII = __builtin_amdgcn_wmma_f32_16x16x4_f32(false, ai1, false, ai1, (short)0, accII, false, false);
    accIR = __builtin_amdgcn_wmma_f32_16x16x4_f32(false, ai1, false, ar1, (short)0, accIR, false, false);
    accRI = __builtin_amdgcn_wmma_f32_16x16x4_f32(false, ar1, false, ai1, (short)0, accRI, false, false);
    accRR = __builtin_amdgcn_wmma_f32_16x16x4_f32(false, ar2, false, ar2, (short)0, accRR, false, false);
    accII = __builtin_amdgcn_wmma_f32_16x16x4_f32(false, ai2, false, ai2, (short)0, accII, false, false);
    accIR = __builtin_amdgcn_wmma_f32_16x16x4_f32(false, ai2, false, ar2, (short)0, accIR, false, false);
    accRI = __builtin_amdgcn_wmma_f32_16x16x4_f32(false, ar2, false, ai2, (short)0, accRI, false, false);
    accRR = __builtin_amdgcn_wmma_f32_16x16x4_f32(false, ar3, false, ar3, (short)0, accRR, false, false);
    accII = __builtin_amdgcn_wmma_f32_16x16x4_f32(false, ai3, false, ai3, (short)0, accII, false, false);
    accIR = __builtin_amdgcn_wmma_f32_16x16x4_f32(false, ai3, false, ar3, (short)0, accIR, false, false);
    accRI = __builtin_amdgcn_wmma_f32_16x16x4_f32(false, ar3, false, ai3, (short)0, accRI, false, false);
  }
  v8f reAcc = accRR + accII;   // Re(A A^H)
  v8f imAcc = accIR - accRI;   // Im(A A^H)
  int ibase = (lane >> 4) * 8; // C/D layout: VGPR v -> M = v or v+8
  int j = lane & 15;
  for (int v = 0; v < 8; v++) {
    int i = ibase + v;
    atomicAdd(&rhoRe[slice * 256 + i * 16 + j], reAcc[v]);
    atomicAdd(&rhoIm[slice * 256 + i * 16 + j], imAcc[v]);
  }
}

// ---------------------------------------------------------------------------
// Kernel E: partial traces -> expectation values -> head matmul -> normalize.
// ---------------------------------------------------------------------------
__global__ void k_head(const float* __restrict__ rhoRe, const float* __restrict__ rhoIm,
                       const float2* __restrict__ heis,
                       const float* __restrict__ headw, const float* __restrict__ headb,
                       float* __restrict__ out) {
  __shared__ float ev[8][4];
  __shared__ float nrm[8];
  int t = threadIdx.x;
  if (t < 32) ((float*)ev)[t] = 0.f;
  if (t < 8) nrm[t] = 0.f;
  __syncthreads();
  {
    int b = t >> 6, idx = t & 63, i = idx >> 3, j = idx & 7;
    auto rd = [&](int slice, int r, int c) -> float2 {
      int o = slice * 256 + r * 16 + c;
      return make_float2(rhoRe[o], rhoIm[o]);
    };
    {  // e1: tr(H1, rho012[b,0])
      float2 O = heis[1 * 64 + i * 8 + j];
      float2 r = make_float2(0.f, 0.f);
      for (int xx = 0; xx < 2; xx++) { float2 v = rd(b * 4 + 0, j * 2 + xx, i * 2 + xx); r.x += v.x; r.y += v.y; }
      atomicAdd(&ev[b][1], O.x * r.x - O.y * r.y);
    }
    {  // e2: tr(H2, rho123[b,0])
      float2 O = heis[2 * 64 + i * 8 + j];
      float2 r = make_float2(0.f, 0.f);
      for (int xx = 0; xx < 2; xx++) { float2 v = rd(b * 4 + 0, xx * 8 + j, xx * 8 + i); r.x += v.x; r.y += v.y; }
      atomicAdd(&ev[b][2], O.x * r.x - O.y * r.y);
    }
    {  // e0: tr(H0, rho_q3[b,3] (x) rho01[b,0])
      float2 O = heis[0 * 64 + i * 8 + j];
      float2 rq = make_float2(0.f, 0.f);
      for (int m = 0; m < 8; m++) { float2 v = rd(b * 4 + 3, m * 2 + (j >> 2), m * 2 + (i >> 2)); rq.x += v.x; rq.y += v.y; }
      float2 r01 = make_float2(0.f, 0.f);
      for (int m = 0; m < 4; m++) { float2 v = rd(b * 4 + 0, (j & 3) * 4 + m, (i & 3) * 4 + m); r01.x += v.x; r01.y += v.y; }
      float2 r = cmul(rq, r01);
      atomicAdd(&ev[b][0], O.x * r.x - O.y * r.y);
    }
    {  // e3: tr(H3, rho23[b,0] (x) rho_q0[b,1])
      float2 O = heis[3 * 64 + i * 8 + j];
      float2 r23 = make_float2(0.f, 0.f);
      for (int m = 0; m < 4; m++) {
        int xx = m >> 1, yy = m & 1;
        float2 v = rd(b * 4 + 0, xx * 8 + yy * 4 + (j >> 1), xx * 8 + yy * 4 + (i >> 1));
        r23.x += v.x; r23.y += v.y;
      }
      float2 rq0 = make_float2(0.f, 0.f);
      for (int m = 0; m < 8; m++) { float2 v = rd(b * 4 + 1, (j & 1) * 8 + m, (i & 1) * 8 + m); rq0.x += v.x; rq0.y += v.y; }
      float2 r = cmul(r23, rq0);
      atomicAdd(&ev[b][3], O.x * r.x - O.y * r.y);
    }
  }
  __syncthreads();
  float val[8];
  for (int b = 0; b < 8; b++) {
    float v = headb[t];
    for (int c = 0; c < 4; c++) v += ev[b][c] * headw[t * 4 + c];
    val[b] = v;
    atomicAdd(&nrm[b], v * v);
  }
  __syncthreads();
  for (int b = 0; b < 8; b++) {
    out[b * 512 + t] = val[b] / fmaxf(sqrtf(nrm[b]), 1e-12f);
  }
}

// ---------------------------------------------------------------------------
extern "C" void kernel_launch(void* const* d_in, const int* in_sizes, int n_in,
                              void* d_out, int out_size, void* d_ws, size_t ws_size,
                              hipStream_t stream) {
  (void)in_sizes; (void)n_in; (void)out_size;
  const float* x     = (const float*)d_in[0];
  const float* posw  = (const float*)d_in[1];
  const float* l1w   = (const float*)d_in[2];
  const float* l2w   = (const float*)d_in[3];
  const float* headw = (const float*)d_in[4];
  const float* headb = (const float*)d_in[5];
  float* out = (float*)d_out;
  char* ws = (char*)d_ws;

  float2* states = (float2*)(ws);              // 10,240 B
  float2* heis   = (float2*)(ws + 16384);      //  2,048 B
  float*  rhoRe  = (float*)(ws + 20480);       // 32,768 B
  float*  rhoIm  = (float*)(ws + 53248);       // 32,768 B (contiguous w/ rhoRe)
  const size_t PSI_OFF = 98304;
  const size_t perSlice = (size_t)NN * 2 * sizeof(float);  // 8 MB (re+im)

  int chunk = 1;
  if (ws_size > PSI_OFF + perSlice) {
    size_t c = (ws_size - PSI_OFF) / perSlice;
    chunk = (int)(c > 32 ? 32 : c);
    if (chunk < 1) chunk = 1;
  }
  float* psiRe = (float*)(ws + PSI_OFF);
  float* psiIm = psiRe + (size_t)chunk * NN;

  k_setup<<<1, 640, 0, stream>>>(x, posw, l2w, states, heis);
  k_zero<<<64, 256, 0, stream>>>(rhoRe, 16384);  // zeros rhoRe+rhoIm

  for (int s0 = 0; s0 < 32; s0 += chunk) {
    int c = (32 - s0 < chunk) ? (32 - s0) : chunk;
    k_psi_lo<<<dim3(256, c), 256, 0, stream>>>(states, l1w, psiRe, psiIm, s0);
    k_wht_hi<<<dim3(128, c), 256, 0, stream>>>(psiRe, psiIm);
    k_rho<<<dim3(16, c), 256, 0, stream>>>(psiRe, psiIm, rhoRe, rhoIm, s0);
  }
  k_head<<<1, 512, 0, stream>>>(rhoRe, rhoIm, heis, headw, headb, out);
}